// DeepRecurrentNeuralNetwork_47287589929865
// MI455X (gfx1250) — compile-verified
//
#include <hip/hip_runtime.h>
#include <math.h>
#include <stdint.h>

typedef __attribute__((ext_vector_type(2))) float v2f;
typedef __attribute__((ext_vector_type(8))) float v8f;

#define BB 128   // batch
#define SS 512   // sequence length
#define II 256   // input features
#define HH 512   // hidden size
#define OO 128   // output features
#define KC 64    // K-chunk staged in LDS per double-buffer slot
#define ZST 68   // LDS row stride in dwords (16B-aligned, bank-conflict-free)

__device__ __forceinline__ float sigmoidf_(float v) {
    return 1.0f / (1.0f + expf(-v));
}

// CDNA5 async copy: each lane copies 16 bytes from the global address gaddr to the LDS address lds_off, tracked by ASYNCcnt.
__device__ __forceinline__ void async_b128(uint32_t lds_off, const float* gaddr) {
    asm volatile("global_load_async_to_lds_b128 %0, %1, off"
                 :: "v"(lds_off), "v"(gaddr)
                 : "memory");
}
__device__ __forceinline__ void wait_async_le4() {
    asm volatile("s_wait_asynccnt 0x4" ::: "memory");
}
__device__ __forceinline__ void wait_async_0() {
    asm volatile("s_wait_asynccnt 0x0" ::: "memory");
}

__global__ void zero_ws_kernel(float* __restrict__ p, int n) {
    int i = blockIdx.x * blockDim.x + threadIdx.x;
    if (i < n) p[i] = 0.0f;
}

// One launch per pipeline diagonal tau; blockIdx.z = layer l, computing t = tau - l.
// Grid: (HH/16, BB/32, 3), block: 128 threads = 4 wave32s (one gate each),
// each wave computes TWO 16x16 M-tiles (32 batch rows) of its gate.
// hbuf: [3][2][BB][HH] (double-buffered by tau parity), cbuf: [3][BB][HH].
__global__ __launch_bounds__(128)
void lstm_step_kernel(const float* __restrict__ x,
                      const float* __restrict__ Wih0, const float* __restrict__ Whh0,
                      const float* __restrict__ bih0, const float* __restrict__ bhh0,
                      const float* __restrict__ Wih1, const float* __restrict__ Whh1,
                      const float* __restrict__ bih1, const float* __restrict__ bhh1,
                      const float* __restrict__ Wih2, const float* __restrict__ Whh2,
                      const float* __restrict__ bih2, const float* __restrict__ bhh2,
                      float* __restrict__ hbuf, float* __restrict__ cbuf, int tau)
{
    const int l = blockIdx.z;
    const int t = tau - l;
    if (t < 0 || t >= SS) return;          // uniform per block

    const int slotW = tau & 1;
    const int slotR = slotW ^ 1;
    const int j0   = blockIdx.x * 16;      // hidden-unit tile
    const int b0   = blockIdx.y * 32;      // batch supertile (2 M-tiles)
    const int tid  = threadIdx.x;
    const int wave = tid >> 5;             // gate: 0=i 1=f 2=g 3=o
    const int lane = tid & 31;
    const int m    = lane & 15;
    const int kh   = (lane >> 4) * 2;
    const int hi   = lane >> 4;

    const int Kin = (l == 0) ? II : HH;
    const float* Wih = (l == 0) ? Wih0 : (l == 1) ? Wih1 : Wih2;
    const float* Whh = (l == 0) ? Whh0 : (l == 1) ? Whh1 : Whh2;
    const float* bih = (l == 0) ? bih0 : (l == 1) ? bih1 : bih2;
    const float* bhh = (l == 0) ? bhh0 : (l == 1) ? bhh1 : bhh2;

    const int n0 = wave * HH + j0;
    const float* wih_row = Wih + (size_t)(n0 + m) * Kin;
    const float* whh_row = Whh + (size_t)(n0 + m) * HH;

    const float* hin = (l == 0) ? (const float*)0
                                : hbuf + ((size_t)(l - 1) * 2 + slotR) * BB * HH;
    const float* hpv = hbuf + ((size_t)l * 2 + slotR) * BB * HH;

    __shared__ float zbuf[2][32][ZST];     // staged activation panel (double buffer)
    __shared__ float gtile[4][32][16];     // gate exchange for cell update
    const uint32_t zbase = (uint32_t)(uintptr_t)&zbuf[0][0][0];

    // Per-thread staging assignment: 512 x b128 transfers per chunk, 4 per thread.
    int rows[4], cdw[4];
    #pragma unroll
    for (int i = 0; i < 4; ++i) {
        const int idx = tid + 128 * i;
        rows[i] = idx >> 4;                // 0..31
        cdw[i]  = (idx & 15) * 4;          // dword column within chunk
    }

    v8f acc0 = {}, acc1 = {};

    #pragma unroll
    for (int ph = 0; ph < 2; ++ph) {
        const int ncols = ph ? HH : Kin;
        const float* wrow = (ph ? whh_row : wih_row) + kh;

        // Per-thread source row pointers for this phase.
        const float* rb[4];
        #pragma unroll
        for (int i = 0; i < 4; ++i) {
            const int r = rows[i];
            const float* src =
                ph ? (hpv + (size_t)(b0 + r) * HH)
                   : ((l == 0) ? x + ((size_t)(b0 + r) * SS + t) * II
                               : hin + (size_t)(b0 + r) * HH);
            rb[i] = src + cdw[i];
        }
        uint32_t lo[2][4];
        #pragma unroll
        for (int bf = 0; bf < 2; ++bf)
            #pragma unroll
            for (int i = 0; i < 4; ++i)
                lo[bf][i] = zbase + (uint32_t)(((bf * 32 + rows[i]) * ZST + cdw[i]) * 4);

        const int nch = ncols / KC;
        #pragma unroll
        for (int i = 0; i < 4; ++i) async_b128(lo[0][i], rb[i]);

        for (int ci = 0; ci < nch; ++ci) {
            const int pb = ci & 1;
            if (ci + 1 < nch) {
                #pragma unroll
                for (int i = 0; i < 4; ++i)
                    async_b128(lo[pb ^ 1][i], rb[i] + (ci + 1) * KC);
                wait_async_le4();          // current chunk landed (in-order)
            } else {
                wait_async_0();
            }
            __syncthreads();

            const float* wr  = wrow + ci * KC;
            const float* za0 = &zbuf[pb][m][kh];
            const float* za1 = &zbuf[pb][16 + m][kh];
            __builtin_prefetch(wr + KC, 0, 3);   // WGP-scope prefetch of next weight chunk
            #pragma unroll 8
            for (int k = 0; k < KC; k += 4) {
                v2f b  = *(const v2f*)(wr + k);
                v2f a0 = *(const v2f*)(za0 + k);
                v2f a1 = *(const v2f*)(za1 + k);
                acc0 = __builtin_amdgcn_wmma_f32_16x16x4_f32(false, a0, false, b,
                                                             (short)0, acc0, false, false);
                acc1 = __builtin_amdgcn_wmma_f32_16x16x4_f32(false, a1, false, b,
                                                             (short)0, acc1, false, false);
            }
            __syncthreads();               // protect pb before next prefetch overwrites it
        }
    }

    // Exchange gate tiles through LDS, then fused LSTM cell update.
    #pragma unroll
    for (int r = 0; r < 8; ++r) {
        gtile[wave][r + hi * 8][m]      = acc0[r];
        gtile[wave][16 + r + hi * 8][m] = acc1[r];
    }
    __syncthreads();

    for (int e = tid; e < 512; e += 128) {
        const int M = e >> 4;              // 0..31 batch row within supertile
        const int N = e & 15;
        const int j = j0 + N;
        const float gi = gtile[0][M][N] + bih[j]          + bhh[j];
        const float gf = gtile[1][M][N] + bih[HH + j]     + bhh[HH + j];
        const float gg = gtile[2][M][N] + bih[2 * HH + j] + bhh[2 * HH + j];
        const float go = gtile[3][M][N] + bih[3 * HH + j] + bhh[3 * HH + j];
        const size_t ci = ((size_t)l * BB + (b0 + M)) * HH + j;
        const float cn = sigmoidf_(gf) * cbuf[ci] + sigmoidf_(gi) * tanhf(gg);
        const float hn = sigmoidf_(go) * tanhf(cn);
        cbuf[ci] = cn;
        hbuf[(((size_t)l * 2 + slotW) * BB + (b0 + M)) * HH + j] = hn;
    }
}

// out[b][o] = h2[b] . fcw[o] + fcb[o]; grid (OO/16, BB/16), one wave32 per block.
__global__ __launch_bounds__(32)
void fc_kernel(const float* __restrict__ h2, const float* __restrict__ fcw,
               const float* __restrict__ fcb, float* __restrict__ out)
{
    const int o0 = blockIdx.x * 16;
    const int b0 = blockIdx.y * 16;
    const int lane = threadIdx.x & 31;
    const int m  = lane & 15;
    const int kh = (lane >> 4) * 2;

    const float* arow = h2  + (size_t)(b0 + m) * HH;
    const float* brow = fcw + (size_t)(o0 + m) * HH;

    v8f acc = {};
    #pragma unroll 8
    for (int k = 0; k < HH; k += 4) {
        v2f a = *(const v2f*)(arow + k + kh);
        v2f b = *(const v2f*)(brow + k + kh);
        acc = __builtin_amdgcn_wmma_f32_16x16x4_f32(false, a, false, b,
                                                    (short)0, acc, false, false);
    }

    const int hi = lane >> 4;
    const float bias = fcb[o0 + m];
    #pragma unroll
    for (int r = 0; r < 8; ++r) {
        const int M = r + hi * 8;
        out[(size_t)(b0 + M) * OO + (o0 + m)] = acc[r] + bias;
    }
}

extern "C" void kernel_launch(void* const* d_in, const int* in_sizes, int n_in,
                              void* d_out, int out_size, void* d_ws, size_t ws_size,
                              hipStream_t stream) {
    (void)in_sizes; (void)n_in; (void)out_size; (void)ws_size;
    const float* x    = (const float*)d_in[0];
    const float* Wih0 = (const float*)d_in[1];
    const float* Whh0 = (const float*)d_in[2];
    const float* bih0 = (const float*)d_in[3];
    const float* bhh0 = (const float*)d_in[4];
    const float* Wih1 = (const float*)d_in[5];
    const float* Whh1 = (const float*)d_in[6];
    const float* bih1 = (const float*)d_in[7];
    const float* bhh1 = (const float*)d_in[8];
    const float* Wih2 = (const float*)d_in[9];
    const float* Whh2 = (const float*)d_in[10];
    const float* bih2 = (const float*)d_in[11];
    const float* bhh2 = (const float*)d_in[12];
    const float* fcw  = (const float*)d_in[13];
    const float* fcb  = (const float*)d_in[14];

    float* hbuf = (float*)d_ws;                       // [3][2][BB][HH]
    float* cbuf = hbuf + (size_t)3 * 2 * BB * HH;     // [3][BB][HH]
    const int state_elems = 9 * BB * HH;

    zero_ws_kernel<<<(state_elems + 255) / 256, 256, 0, stream>>>(hbuf, state_elems);

    for (int tau = 0; tau <= SS + 1; ++tau) {
        lstm_step_kernel<<<dim3(HH / 16, BB / 32, 3), 128, 0, stream>>>(
            x, Wih0, Whh0, bih0, bhh0, Wih1, Whh1, bih1, bhh1,
            Wih2, Whh2, bih2, bhh2, hbuf, cbuf, tau);
    }

    const int fslot = (SS + 1) & 1;
    const float* h2 = hbuf + ((size_t)2 * 2 + fslot) * BB * HH;
    fc_kernel<<<dim3(OO / 16, BB / 16), 32, 0, stream>>>(h2, fcw, fcb, (float*)d_out);
}